// SemiMarkovModule_30597347016780
// MI455X (gfx1250) — compile-verified
//
#include <hip/hip_runtime.h>
#include <math.h>

#define NEGV (-1.0e9f)

typedef float v2f __attribute__((ext_vector_type(2)));
typedef float v8f __attribute__((ext_vector_type(8)));

// workspace layout (float offsets)
#define WS_PREC   0        // 64*64 precision matrix
#define WS_PMU    4096     // 32*64  Pmu[c][d] = (P @ mu_c)[d]
#define WS_MUPMU  6144     // 32     mu_c^T P mu_c
#define WS_LS     6176     // 20*32  duration scores ls[k][c] (d = k+1)
#define WS_E      6816     // 32*32  exp(log_softmax(trans, axis=to))[to][from]
#define WS_INIT   7840     // 32     log_softmax(init_logits)
#define WS_CONST  7872     // 1      D*log(2*pi) + logdet
#define WS_EMIT   8192     // B*T*32 emission log-probs

static constexpr int DDIM = 64;
static constexpr int CCLS = 32;
static constexpr int KDUR = 20;

// ---------------------------------------------------------------------------
// Prep: invert cov (Gauss-Jordan in LDS), logdet, Pmu, muPmu, durations,
// exp-normalized transition matrix, init log-softmax.
// ---------------------------------------------------------------------------
__global__ void prep_kernel(const float* __restrict__ cov,
                            const float* __restrict__ mu,
                            const float* __restrict__ plr,
                            const float* __restrict__ tl,
                            const float* __restrict__ il,
                            float* __restrict__ wsF) {
  __shared__ float aug[64 * 128];   // [A | I], 32 KB
  __shared__ float pivrow[128];
  __shared__ float fac[64];
  __shared__ float colLSE[32];
  __shared__ float ld_acc;
  const int tid = threadIdx.x;

  for (int i = tid; i < 64 * 128; i += 256) {
    int r = i >> 7, c = i & 127;
    aug[i] = (c < 64) ? cov[r * 64 + c] : ((c - 64 == r) ? 1.f : 0.f);
  }
  if (tid == 0) ld_acc = 0.f;
  __syncthreads();

  for (int p = 0; p < 64; ++p) {
    float pv = aug[p * 128 + p];
    if (tid == 0) ld_acc += __logf(fabsf(pv));
    if (tid < 128) pivrow[tid] = aug[p * 128 + tid] / pv;
    if (tid < 64)  fac[tid] = (tid == p) ? 0.f : aug[tid * 128 + p];
    __syncthreads();
    if (tid < 128) aug[p * 128 + tid] = pivrow[tid];
    for (int i = tid; i < 64 * 128; i += 256) {
      int r = i >> 7, c = i & 127;
      if (r != p) aug[i] -= fac[r] * pivrow[c];
    }
    __syncthreads();
  }

  // precision matrix -> ws
  for (int i = tid; i < 64 * 64; i += 256) {
    int r = i >> 6, c = i & 63;
    wsF[WS_PREC + i] = aug[r * 128 + 64 + c];
  }
  // Pmu[c][d] = sum_e P[d][e] * mu[c][e]
  for (int idx = tid; idx < 32 * 64; idx += 256) {
    int c = idx >> 6, dd = idx & 63;
    float s = 0.f;
    for (int e = 0; e < 64; ++e) s += aug[dd * 128 + 64 + e] * mu[c * 64 + e];
    wsF[WS_PMU + c * 64 + dd] = s;
  }
  // muPmu[c] (recomputed from LDS to avoid global visibility ordering)
  if (tid < 32) {
    float s = 0.f;
    for (int dd = 0; dd < 64; ++dd) {
      float t = 0.f;
      for (int e = 0; e < 64; ++e) t += aug[dd * 128 + 64 + e] * mu[tid * 64 + e];
      s += mu[tid * 64 + dd] * t;
    }
    wsF[WS_MUPMU + tid] = s;
  }
  // Poisson duration scores, d = k+1
  for (int idx = tid; idx < KDUR * 32; idx += 256) {
    int k = idx >> 5, c = idx & 31;
    float lr = plr[c];
    wsF[WS_LS + idx] = (float)(k + 1) * lr - __expf(lr) - lgammaf((float)(k + 2));
  }
  // transition: log_softmax over 'to' (axis 0), then exponentiate
  if (tid < 32) {
    float m = NEGV;
    for (int to = 0; to < 32; ++to) m = fmaxf(m, tl[to * 32 + tid]);
    float s = 0.f;
    for (int to = 0; to < 32; ++to) s += __expf(tl[to * 32 + tid] - m);
    colLSE[tid] = m + __logf(s);
  }
  __syncthreads();
  for (int idx = tid; idx < 32 * 32; idx += 256) {
    int to = idx >> 5, from = idx & 31;
    wsF[WS_E + idx] = __expf(tl[to * 32 + from] - colLSE[from]);
  }
  // init log-softmax
  if (tid < 32) {
    float m = NEGV;
    for (int c = 0; c < 32; ++c) m = fmaxf(m, il[c]);
    float s = 0.f;
    for (int c = 0; c < 32; ++c) s += __expf(il[c] - m);
    wsF[WS_INIT + tid] = il[tid] - (m + __logf(s));
  }
  if (tid == 0) wsF[WS_CONST] = 64.f * 1.8378770664093453f + ld_acc; // D*log(2pi)+logdet
}

// ---------------------------------------------------------------------------
// Emission: one wave per 16-row tile of BT. WMMA f32 16x16x4:
//   Y    = X @ P        (4 N-tiles)   -> quad = rowdot(Y, X)
//   cross= X @ Pmu^T    (2 N-tiles)
// P symmetric and Pmu row-major => both B operands are contiguous float2.
// emit[m][c] = -0.5*(quad[m] - 2*cross[m][c] + muPmu[c] + const)
// ---------------------------------------------------------------------------
__global__ void emission_kernel(const float* __restrict__ x,
                                const float* __restrict__ wsF,
                                float* __restrict__ emit) {
  const int lane = threadIdx.x;
  const int m0 = blockIdx.x * 16;
  const int mrow = lane & 15;
  const int hi = lane >> 4;
  const float* prec = wsF + WS_PREC;
  const float* pmu  = wsF + WS_PMU;

  v8f accY0 = {}, accY1 = {}, accY2 = {}, accY3 = {};
  v8f accC0 = {}, accC1 = {};

#pragma unroll
  for (int kk = 0; kk < 16; ++kk) {
    const int koff = kk * 4 + hi * 2;
    v2f a   = *(const v2f*)(x    + (m0 + mrow) * 64 + koff);
    v2f bp0 = *(const v2f*)(prec + (0  + mrow) * 64 + koff);
    v2f bp1 = *(const v2f*)(prec + (16 + mrow) * 64 + koff);
    v2f bp2 = *(const v2f*)(prec + (32 + mrow) * 64 + koff);
    v2f bp3 = *(const v2f*)(prec + (48 + mrow) * 64 + koff);
    v2f bm0 = *(const v2f*)(pmu  + (0  + mrow) * 64 + koff);
    v2f bm1 = *(const v2f*)(pmu  + (16 + mrow) * 64 + koff);
    accY0 = __builtin_amdgcn_wmma_f32_16x16x4_f32(false, a, false, bp0, (short)0, accY0, false, false);
    accY1 = __builtin_amdgcn_wmma_f32_16x16x4_f32(false, a, false, bp1, (short)0, accY1, false, false);
    accY2 = __builtin_amdgcn_wmma_f32_16x16x4_f32(false, a, false, bp2, (short)0, accY2, false, false);
    accY3 = __builtin_amdgcn_wmma_f32_16x16x4_f32(false, a, false, bp3, (short)0, accY3, false, false);
    accC0 = __builtin_amdgcn_wmma_f32_16x16x4_f32(false, a, false, bm0, (short)0, accC0, false, false);
    accC1 = __builtin_amdgcn_wmma_f32_16x16x4_f32(false, a, false, bm1, (short)0, accC1, false, false);
  }

  // quad partials from D-layout: element (m = v + 8*hi, d = nt*16 + mrow)
  float qp[8];
#pragma unroll
  for (int v = 0; v < 8; ++v) {
    const int rb = (m0 + v + 8 * hi) * 64 + mrow;
    qp[v] = accY0[v] * x[rb] + accY1[v] * x[rb + 16] +
            accY2[v] * x[rb + 32] + accY3[v] * x[rb + 48];
  }
#pragma unroll
  for (int v = 0; v < 8; ++v) {
#pragma unroll
    for (int mask = 1; mask <= 8; mask <<= 1)
      qp[v] += __shfl_xor(qp[v], mask, 32);   // reduce within 16-lane group
  }

  const float mc0 = wsF[WS_MUPMU + mrow];
  const float mc1 = wsF[WS_MUPMU + 16 + mrow];
  const float cst = wsF[WS_CONST];
#pragma unroll
  for (int v = 0; v < 8; ++v) {
    const int m = m0 + v + 8 * hi;
    emit[m * 32 + mrow]      = -0.5f * (qp[v] - 2.f * accC0[v] + mc0 + cst);
    emit[m * 32 + 16 + mrow] = -0.5f * (qp[v] - 2.f * accC1[v] + mc1 + cst);
  }
}

// ---------------------------------------------------------------------------
// Scan: one wave per sequence b, lane = class c. All recursion state in
// registers; emission rows software-pipelined 4 deep; no barriers.
// ---------------------------------------------------------------------------
__global__ void scan_kernel(const float* __restrict__ emit,
                            const float* __restrict__ wsF,
                            const int* __restrict__ lengths,
                            float* __restrict__ out, int T) {
  const int b = blockIdx.x;
  const int lane = threadIdx.x;

  float lsr[KDUR];
#pragma unroll
  for (int k = 0; k < KDUR; ++k) lsr[k] = wsF[WS_LS + k * 32 + lane];
  float Erow[32];
#pragma unroll
  for (int f = 0; f < 32; ++f) Erow[f] = wsF[WS_E + lane * 32 + f];
  const int len = lengths[b];

  float ast[KDUR], er[KDUR];
#pragma unroll
  for (int i = 0; i < KDUR; ++i) { ast[i] = NEGV; er[i] = 0.f; }
  ast[0] = wsF[WS_INIT + lane];            // astar[-1] = init (== buf0[0])

  const float* eb = emit + (size_t)b * T * 32 + lane;
  float pend[4];
#pragma unroll
  for (int u = 0; u < 4; ++u) pend[u] = eb[(size_t)u * 32];

  for (int t = 0; t < T; t += 4) {
#pragma unroll
    for (int u = 0; u < 4; ++u) {
      const int tt = t + u;
      const float e = pend[u];
      int pf = tt + 4; if (pf > T - 1) pf = T - 1;
      pend[u] = eb[(size_t)pf * 32];       // prefetch 4 steps ahead

      // push emission into span ring: er[i] = emit[tt - i]
#pragma unroll
      for (int i = KDUR - 1; i > 0; --i) er[i] = er[i - 1];
      er[0] = e;

      // alpha[c] = LSE_d( astar[t-d] + span_d + ls[d] )
      float s = 0.f, m = NEGV;
      float vv[KDUR];
#pragma unroll
      for (int d = 0; d < KDUR; ++d) {
        s += er[d];
        vv[d] = ast[d] + s + lsr[d];
        m = fmaxf(m, vv[d]);
      }
      float acc = 0.f;
#pragma unroll
      for (int d = 0; d < KDUR; ++d) acc += __expf(vv[d] - m);
      const float alpha = m + __logf(acc);

      // wave max over classes
      float wm = alpha;
#pragma unroll
      for (int mk = 16; mk >= 1; mk >>= 1) wm = fmaxf(wm, __shfl_xor(wm, mk, 32));
      const float p = __expf(alpha - wm);

      if (tt == len - 1) {                 // uniform per wave
        float ps = p;
#pragma unroll
        for (int mk = 16; mk >= 1; mk >>= 1) ps += __shfl_xor(ps, mk, 32);
        if (lane == 0) out[b] = -(wm + __logf(ps));
        return;
      }

      // astar[to] = wm + log( sum_from p[from] * E[to][from] )
      float sa = 0.f;
#pragma unroll
      for (int f = 0; f < 32; ++f) sa = fmaf(__shfl(p, f, 32), Erow[f], sa);
      const float astar = wm + __logf(sa);

#pragma unroll
      for (int i = KDUR - 1; i > 0; --i) ast[i] = ast[i - 1];
      ast[0] = astar;
    }
  }
}

// ---------------------------------------------------------------------------
extern "C" void kernel_launch(void* const* d_in, const int* in_sizes, int n_in,
                              void* d_out, int out_size, void* d_ws, size_t ws_size,
                              hipStream_t stream) {
  const float* features = (const float*)d_in[0];
  const float* plr      = (const float*)d_in[1];
  const float* mu       = (const float*)d_in[2];
  const float* cov      = (const float*)d_in[3];
  const float* tl       = (const float*)d_in[4];
  const float* il       = (const float*)d_in[5];
  const int*   lengths  = (const int*)d_in[6];

  const int B = in_sizes[6];
  const int T = in_sizes[0] / (B * DDIM);

  float* wsF  = (float*)d_ws;
  float* emit = wsF + WS_EMIT;
  float* out  = (float*)d_out;

  prep_kernel<<<dim3(1), dim3(256), 0, stream>>>(cov, mu, plr, tl, il, wsF);
  emission_kernel<<<dim3((B * T) / 16), dim3(32), 0, stream>>>(features, wsF, emit);
  scan_kernel<<<dim3(B), dim3(32), 0, stream>>>(emit, wsF, lengths, out, T);
}